// PostprocessingYoloV8Pose_69784628625815
// MI455X (gfx1250) — compile-verified
//
#include <hip/hip_runtime.h>
#include <hip/hip_bf16.h>
#include <stdint.h>

#define BS       16
#define NCH      56
#define N_ANCH   33600
#define NPAD     65536          // next pow2 >= N_ANCH, for bitonic sort
#define CHUNK    4096           // LDS-resident bitonic chunk (32 KB of 320 KB WGP LDS)
#define MAX_NMS  3000
#define MAX_DET  300
#define CONF_T   0.25f
#define IOU_T    0.7f
#define IMGW     1280.0f
#define IMGH     1280.0f

// ---- CDNA5 async LDS-DMA helpers (ASYNCcnt path) ---------------------------
__device__ __forceinline__ void async_ld_b32(unsigned lds_off, unsigned long long gaddr) {
    asm volatile("global_load_async_to_lds_b32 %0, %1, off"
                 :: "v"(lds_off), "v"(gaddr) : "memory");
}
__device__ __forceinline__ void async_ld_b64(unsigned lds_off, unsigned long long gaddr) {
    asm volatile("global_load_async_to_lds_b64 %0, %1, off"
                 :: "v"(lds_off), "v"(gaddr) : "memory");
}
__device__ __forceinline__ void async_st_b64(unsigned long long gaddr, unsigned lds_off) {
    asm volatile("global_store_async_from_lds_b64 %0, %1, off"
                 :: "v"(gaddr), "v"(lds_off) : "memory");
}
__device__ __forceinline__ void wait_async0() {
#if __has_builtin(__builtin_amdgcn_s_wait_asynccnt)
    __builtin_amdgcn_s_wait_asynccnt(0);
#else
    asm volatile("s_wait_asynccnt 0" ::: "memory");
#endif
}

// ---------------------------------------------------------------------------
// LDS-resident bitonic full sort of aligned 4096-key chunks: stages k=2..4096,
// all passes in LDS. Chunks are staged in and written back via the async
// global<->LDS DMA path. Descending order; direction from the chunk's global
// position within its image. Replaces 78 global passes with 1 launch.
// ---------------------------------------------------------------------------
__global__ __launch_bounds__(1024)
void bitonic_local_sort(unsigned long long* __restrict__ keys) {
    const int tid = threadIdx.x;
    __shared__ unsigned long long s_k[CHUNK];               // 32 KB
    const size_t base = (size_t)blockIdx.x * CHUNK;         // grid = BS*NPAD/CHUNK
    const int gbase = (int)(base & (size_t)(NPAD - 1));     // index within image

    for (int t = tid; t < CHUNK; t += 1024)
        async_ld_b64((unsigned)(size_t)&s_k[t],
                     (unsigned long long)(size_t)&keys[base + t]);
    wait_async0();
    __syncthreads();

    for (int k = 2; k <= CHUNK; k <<= 1) {
        for (int j = k >> 1; j >= 1; j >>= 1) {
            for (int t = tid; t < CHUNK / 2; t += 1024) {
                int i = (t << 1) - (t & (j - 1));           // pair base
                int p = i + j;
                bool descRegion = (((gbase + i) & k) == 0);
                unsigned long long a = s_k[i], c = s_k[p];
                bool doSwap = descRegion ? (a < c) : (a > c);
                if (doSwap) { s_k[i] = c; s_k[p] = a; }
            }
            __syncthreads();
        }
    }

    for (int t = tid; t < CHUNK; t += 1024)
        async_st_b64((unsigned long long)(size_t)&keys[base + t],
                     (unsigned)(size_t)&s_k[t]);
    wait_async0();
}

// ---------------------------------------------------------------------------
// LDS-resident merge tail of stage k: passes j = CHUNK/2 .. 1 (partners stay
// inside an aligned CHUNK for j < CHUNK). Async-staged like the local sort.
// ---------------------------------------------------------------------------
__global__ __launch_bounds__(1024)
void bitonic_fused_merge(unsigned long long* __restrict__ keys, int k) {
    const int tid = threadIdx.x;
    __shared__ unsigned long long s_k[CHUNK];
    const size_t base = (size_t)blockIdx.x * CHUNK;
    const int gbase = (int)(base & (size_t)(NPAD - 1));

    for (int t = tid; t < CHUNK; t += 1024)
        async_ld_b64((unsigned)(size_t)&s_k[t],
                     (unsigned long long)(size_t)&keys[base + t]);
    wait_async0();
    __syncthreads();

    for (int j = CHUNK >> 1; j >= 1; j >>= 1) {
        for (int t = tid; t < CHUNK / 2; t += 1024) {
            int i = (t << 1) - (t & (j - 1));
            int p = i + j;
            bool descRegion = (((gbase + i) & k) == 0);
            unsigned long long a = s_k[i], c = s_k[p];
            bool doSwap = descRegion ? (a < c) : (a > c);
            if (doSwap) { s_k[i] = c; s_k[p] = a; }
        }
        __syncthreads();
    }

    for (int t = tid; t < CHUNK; t += 1024)
        async_st_b64((unsigned long long)(size_t)&keys[base + t],
                     (unsigned)(size_t)&s_k[t]);
    wait_async0();
}

// ---------------------------------------------------------------------------
// Kernel: build 64-bit sort keys.
//   hi32 = order-preserving uint encoding of masked score (invalid -> -1.0)
//   lo32 = ~anchor_index (stable tie-break toward smaller index, matching
//          jnp.argsort on descending scores). Padding slots get key 0, which
//   sinks below every real key (min real key = encode(-1.0f) = 0x407FFFFF<<32).
// ---------------------------------------------------------------------------
__global__ void build_keys(const float* __restrict__ preds,
                           unsigned long long* __restrict__ keys) {
    int gid = blockIdx.x * blockDim.x + threadIdx.x;
    int b = gid >> 16;          // NPAD == 1<<16
    int i = gid & (NPAD - 1);
    if (b >= BS) return;
    unsigned long long key = 0ull;
    if (i < N_ANCH) {
        float s = preds[((size_t)b * NCH + 4) * N_ANCH + i];
        float m = (s > CONF_T) ? s : -1.0f;
        unsigned u = __float_as_uint(m);
        u = (u & 0x80000000u) ? ~u : (u | 0x80000000u);   // ascending-orderable
        key = ((unsigned long long)u << 32) | (unsigned)(~(unsigned)i);
    }
    keys[(size_t)b * NPAD + i] = key;
}

// ---------------------------------------------------------------------------
// Global bitonic pass for j >= CHUNK (cross-chunk partners). 8 MB key array
// stays resident in the 192 MB L2 across all passes.
// ---------------------------------------------------------------------------
__global__ void bitonic_step(unsigned long long* __restrict__ keys, int k, int j) {
    int gid = blockIdx.x * blockDim.x + threadIdx.x;
    int b = gid >> 16;
    int i = gid & (NPAD - 1);
    if (b >= BS) return;
    int p = i ^ j;
    if (p <= i) return;
    size_t base = (size_t)b * NPAD;
    unsigned long long a = keys[base + i];
    unsigned long long c = keys[base + p];
    bool descRegion = ((i & k) == 0);
    bool doSwap = descRegion ? (a < c) : (a > c);
    if (doSwap) { keys[base + i] = c; keys[base + p] = a; }
}

// ---------------------------------------------------------------------------
// Per-image NMS + output (one workgroup per image, 32 wave32 waves):
//   async global->LDS gather of top-3000 boxes (strided channels), clip to
//   xyxy in LDS, greedy rank-ordered NMS on the LDS table, compact first 300
//   survivors, zero-fill + scatter outputs (kpts re-gathered from preds).
// ---------------------------------------------------------------------------
__global__ __launch_bounds__(1024)
void nms_output(const float* __restrict__ preds,
                const unsigned long long* __restrict__ keys,
                float* __restrict__ out) {
    const int b   = blockIdx.x;
    const int tid = threadIdx.x;
    const int nthr = blockDim.x;

    __shared__ float    s_box[MAX_NMS * 4];                 // 48000 B (clipped xyxy)
    __shared__ int      s_anchor[MAX_NMS];                  // 12000 B
    __shared__ unsigned s_keep[(MAX_NMS + 31) / 32];        //   376 B
    __shared__ int      s_sel[MAX_DET];                     //  1200 B
    __shared__ int      s_nsel;

    for (int t = tid; t < (MAX_NMS + 31) / 32; t += nthr) s_keep[t] = 0u;
    if (tid == 0) s_nsel = 0;
    __syncthreads();

    const unsigned long long* mykeys = keys + (size_t)b * NPAD;
    const float* img = preds + (size_t)b * NCH * N_ANCH;

    // ---- stage boxes into LDS via async loads; set valid bits from scores
    for (int t = tid; t < MAX_NMS; t += nthr) {
        unsigned long long key = mykeys[t];
        unsigned a = ~((unsigned)key);
        if (a >= (unsigned)N_ANCH) a = 0;   // can't happen in top-3000; safety
        s_anchor[t] = (int)a;
        const float* col = img + a;         // channel c at col[c * N_ANCH]
        unsigned l0 = (unsigned)(size_t)(&s_box[t * 4]);   // low32 of flat = LDS offset
        unsigned long long g0 = (unsigned long long)(size_t)col;
        const unsigned long long cstride = (unsigned long long)N_ANCH * 4ull; // bytes
        async_ld_b32(l0,      g0);
        async_ld_b32(l0 + 4,  g0 + cstride);
        async_ld_b32(l0 + 8,  g0 + 2 * cstride);
        async_ld_b32(l0 + 12, g0 + 3 * cstride);
        float sc = col[(size_t)4 * N_ANCH];
        if (sc > CONF_T) atomicOr(&s_keep[t >> 5], 1u << (t & 31));
    }
    wait_async0();
    __syncthreads();

    // ---- convert xywh -> clipped xyxy in place
    for (int t = tid; t < MAX_NMS; t += nthr) {
        float cx = s_box[t * 4 + 0], cy = s_box[t * 4 + 1];
        float w  = s_box[t * 4 + 2], h  = s_box[t * 4 + 3];
        float x1 = fminf(fmaxf(cx - 0.5f * w, 0.0f), IMGW);
        float y1 = fminf(fmaxf(cy - 0.5f * h, 0.0f), IMGH);
        float x2 = fminf(fmaxf(cx + 0.5f * w, 0.0f), IMGW);
        float y2 = fminf(fmaxf(cy + 0.5f * h, 0.0f), IMGH);
        s_box[t * 4 + 0] = x1; s_box[t * 4 + 1] = y1;
        s_box[t * 4 + 2] = x2; s_box[t * 4 + 3] = y2;
    }

    // ---- greedy NMS: serial over rank i, parallel over j > i
    for (int i = 0; i < MAX_NMS - 1; ++i) {
        __syncthreads();   // publish previous rank's suppressions
        if (!((s_keep[i >> 5] >> (i & 31)) & 1u)) continue;   // uniform branch
        float ax1 = s_box[i * 4 + 0], ay1 = s_box[i * 4 + 1];
        float ax2 = s_box[i * 4 + 2], ay2 = s_box[i * 4 + 3];
        float aarea = (ax2 - ax1) * (ay2 - ay1);
        for (int j = i + 1 + tid; j < MAX_NMS; j += nthr) {
            if (!((s_keep[j >> 5] >> (j & 31)) & 1u)) continue;
            float bx1 = s_box[j * 4 + 0], by1 = s_box[j * 4 + 1];
            float bx2 = s_box[j * 4 + 2], by2 = s_box[j * 4 + 3];
            float iw = fmaxf(fminf(ax2, bx2) - fmaxf(ax1, bx1), 0.0f);
            float ih = fmaxf(fminf(ay2, by2) - fmaxf(ay1, by1), 0.0f);
            float inter = iw * ih;
            float barea = (bx2 - bx1) * (by2 - by1);
            float iou = inter / (aarea + barea - inter + 1e-9f);
            if (iou > IOU_T) atomicAnd(&s_keep[j >> 5], ~(1u << (j & 31)));
        }
    }
    __syncthreads();

    // ---- compact first MAX_DET survivors (rank order == t order)
    if (tid == 0) {
        int cnt = 0;
        for (int t = 0; t < MAX_NMS && cnt < MAX_DET; ++t)
            if ((s_keep[t >> 5] >> (t & 31)) & 1u) s_sel[cnt++] = t;
        s_nsel = cnt;
    }
    __syncthreads();
    int nsel = s_nsel;

    // ---- output slices (flat concat: ob[16,300,4] | osc[16,300,1] | okp[16,300,51])
    float* ob  = out + (size_t)b * MAX_DET * 4;
    float* osc = out + (size_t)BS * MAX_DET * 4 + (size_t)b * MAX_DET;
    float* okp = out + (size_t)BS * MAX_DET * 5 + (size_t)b * MAX_DET * 51;

    for (int t = tid; t < MAX_DET * 4;  t += nthr) ob[t]  = 0.0f;
    for (int t = tid; t < MAX_DET;      t += nthr) osc[t] = 0.0f;
    for (int t = tid; t < MAX_DET * 51; t += nthr) okp[t] = 0.0f;
    __syncthreads();

    for (int w = tid; w < nsel * 4; w += nthr) {
        int r = w >> 2, c = w & 3;
        ob[w] = s_box[s_sel[r] * 4 + c];
    }
    for (int r = tid; r < nsel; r += nthr) {
        int a = s_anchor[s_sel[r]];
        osc[r] = img[(size_t)4 * N_ANCH + a];
    }
    for (int w = tid; w < nsel * 51; w += nthr) {
        int r = w / 51, c = w - r * 51;
        int a = s_anchor[s_sel[r]];
        float v = img[((size_t)5 + c) * N_ANCH + a];
        if (c == 0) v = fminf(fmaxf(v, 0.0f), IMGW);   // reference clips only
        if (c == 1) v = fminf(fmaxf(v, 0.0f), IMGH);   // kpt channels 0 and 1
        okp[w] = v;
    }
}

// ---------------------------------------------------------------------------
extern "C" void kernel_launch(void* const* d_in, const int* in_sizes, int n_in,
                              void* d_out, int out_size, void* d_ws, size_t ws_size,
                              hipStream_t stream) {
    (void)in_sizes; (void)n_in; (void)out_size; (void)ws_size;
    const float* preds = (const float*)d_in[0];
    float* out = (float*)d_out;
    unsigned long long* keys = (unsigned long long*)d_ws;   // 16*65536*8 = 8 MiB

    const int total = BS * NPAD;                 // 1,048,576 elements
    const int gblk = 256, ggrd = total / gblk;   // 4096 blocks (global passes)
    const int cgrd = total / CHUNK;              // 256 chunk blocks (LDS passes)

    build_keys<<<ggrd, gblk, 0, stream>>>(preds, keys);

    // stages k = 2..CHUNK entirely in LDS (1 launch instead of 78)
    bitonic_local_sort<<<cgrd, 1024, 0, stream>>>(keys);

    // stages k = 2*CHUNK..NPAD: cross-chunk passes global, tail fused in LDS
    for (int k = CHUNK * 2; k <= NPAD; k <<= 1) {
        for (int j = k >> 1; j >= CHUNK; j >>= 1)
            bitonic_step<<<ggrd, gblk, 0, stream>>>(keys, k, j);
        bitonic_fused_merge<<<cgrd, 1024, 0, stream>>>(keys, k);
    }

    nms_output<<<BS, 1024, 0, stream>>>(preds, keys, out);
}